// Net_63342177681543
// MI455X (gfx1250) — compile-verified
//
#include <hip/hip_runtime.h>
#include <hip/hip_bf16.h>
#include <math.h>

// ---------------- problem constants (from reference) ----------------
#define BATCH     65536
#define DIN       54
#define NBAS      25      // basis functions
#define KPAD      28      // NBAS padded to multiple of 4 for WMMA K
#define QSTRIDE   30      // Q row stride in ws (even -> 8B aligned pairs)
#define TSTEPS    301     // output timesteps (includes t=0)
#define NSTEP     300     // integration steps
#define NCHUNK    19      // ceil(300/16)
#define QROWS     304     // 19*16 padded rows (rows >=300 are zero)
#define TAU_F     3.0f
#define DT_F      0.01f
#define AZ_F      25.0f

typedef float v2f __attribute__((ext_vector_type(2)));
typedef float v8f __attribute__((ext_vector_type(8)));

// ---------------------------------------------------------------
// Kernel 1: precompute Q[t][n] = psi_n(x_t) * x_t / sum_n psi_n(x_t)
// x_t = r^t with r = 1 - A_X*DT/TAU  (phase BEFORE update at step t+1)
// Rows >= NSTEP are zero so the tail chunk multiplies by zero.
// ---------------------------------------------------------------
__global__ void dmp_q_kernel(float* __restrict__ Q) {
    int i = blockIdx.x * blockDim.x + threadIdx.x;
    if (i >= QROWS) return;
    const float r = 1.0f - DT_F / TAU_F;          // A_X = 1
    float xph = expf((float)i * logf(r));         // r^i
    float psi[NBAS];
    float s = 0.0f;
    #pragma unroll
    for (int n = 0; n < NBAS; ++n) {
        float c  = expf(-(float)n / 24.0f);       // exp(-A_X * n/(N-1))
        float s2 = 125.0f / c;                    // N^1.5 / c / A_X
        float d  = xph - c;
        float p  = expf(-0.5f * d * d / s2);
        psi[n] = p;
        s += p;
    }
    float scale = (i < NSTEP) ? (xph / s) : 0.0f;
    #pragma unroll
    for (int k = 0; k < QSTRIDE; ++k)
        Q[i * QSTRIDE + k] = (k < NBAS) ? psi[k] * scale : 0.0f;
}

// ---------------------------------------------------------------
// Kernel 2: fused input-layer GEMM + DMP rollout.
// 128 threads = 4 waves; each wave owns 16 batches = 32 (batch,dof) rows.
// Forcing term F = Wrows(32x28) . Q_chunk^T via v_wmma_f32_16x16x4_f32,
// oriented so D has M = t (sequential recurrence stays lane-local).
// ---------------------------------------------------------------
__launch_bounds__(128)
__global__ void dmp_main_kernel(const float* __restrict__ x,
                                const float* __restrict__ W,
                                const float* __restrict__ bias,
                                const float* __restrict__ Qg,
                                float* __restrict__ out) {
    __shared__ float xs [4][16 * DIN];   // x tile per wave
    __shared__ float hs [4][16 * DIN];   // h tile per wave
    __shared__ float wm [4][32 * KPAD];  // per-row weights, zero padded
    __shared__ float fxs[4][32 * 17];    // D spill (stride 17: no bank conflicts)
    __shared__ float yst[4][32 * 16];    // y staging for coalesced stores

    const int tid  = threadIdx.x;
    const int wave = tid >> 5;
    const int lane = tid & 31;
    const int hl   = lane >> 4;          // half-wave select
    const int nl   = lane & 15;

    const int batch0 = blockIdx.x * 64 + wave * 16;

    // ---- stage x tile (per-wave, coalesced) ----
    const float* xg = x + (size_t)batch0 * DIN;
    for (int i = lane; i < 16 * DIN; i += 32) xs[wave][i] = xg[i];

    // ---- input layer: h = x W^T + b (scalar; W/b are hot in cache) ----
    for (int o = lane; o < 16 * DIN; o += 32) {
        int lb = o / DIN, j = o - lb * DIN;
        float acc = bias[j];
        const float* xr = &xs[wave][lb * DIN];
        const float* wr = &W[j * DIN];
        #pragma unroll 6
        for (int k = 0; k < DIN; ++k) acc = fmaf(xr[k], wr[k], acc);
        hs[wave][o] = acc;
    }

    // ---- per-row (lane = row) extraction: y0, goal, weights ----
    const int lb  = lane >> 1;
    const int dof = lane & 1;
    float y    = hs[wave][lb * DIN + dof];
    float goal = hs[wave][lb * DIN + 2 + dof];
    float amp  = goal - y;
    float z    = 0.0f;
    #pragma unroll
    for (int k = 0; k < KPAD; ++k)
        wm[wave][lane * KPAD + k] =
            (k < NBAS) ? hs[wave][lb * DIN + 4 + dof * NBAS + k] : 0.0f;

    // t = 0 output is y0
    const int batch = batch0 + lb;
    out[(size_t)batch * (2 * TSTEPS) + dof * TSTEPS] = y;

    // ---- build B fragments (K x N = 4 x 16, N = rows); constant over t ----
    // lanes 0-15: N = nl, K = {4kk, 4kk+1}; lanes 16-31: N = nl, K = {4kk+2, 4kk+3}
    v2f bfr[2][7];
    #pragma unroll
    for (int rt = 0; rt < 2; ++rt) {
        #pragma unroll
        for (int kk = 0; kk < 7; ++kk) {
            int k0  = 4 * kk + 2 * hl;
            int row = rt * 16 + nl;
            v2f b;
            b.x = wm[wave][row * KPAD + k0];
            b.y = wm[wave][row * KPAD + k0 + 1];
            bfr[rt][kk] = b;
        }
    }

    const float dy_c = DT_F / TAU_F;     // y += z * dt/tau
    const float bz_c = AZ_F * 0.25f;     // beta_z

    // ---- rollout: 19 chunks of 16 timesteps ----
    for (int ch = 0; ch < NCHUNK; ++ch) {
        const int base = ch * 16;

        v8f d0, d1;
        #pragma unroll
        for (int i = 0; i < 8; ++i) { d0[i] = 0.0f; d1[i] = 0.0f; }

        // A: M x K = 16 x 4, M = t-local. lanes 0-15: M = nl, K = {4kk,4kk+1};
        // lanes 16-31: M = nl, K = {4kk+2,4kk+3}. Q read from global (L2 hot).
        #pragma unroll
        for (int kk = 0; kk < 7; ++kk) {
            const float* qr = &Qg[(base + nl) * QSTRIDE + 4 * kk + 2 * hl];
            v2f a;
            a.x = qr[0];
            a.y = qr[1];
            d0 = __builtin_amdgcn_wmma_f32_16x16x4_f32(
                     false, a, false, bfr[0][kk], (short)0, d0, false, false);
            d1 = __builtin_amdgcn_wmma_f32_16x16x4_f32(
                     false, a, false, bfr[1][kk], (short)0, d1, false, false);
        }

        // D layout: lane (hl,nl), VGPR v -> element (M = v + 8*hl, N = nl).
        // Spill to fxs[row][t_local], stride 17 (conflict-free).
        #pragma unroll
        for (int v = 0; v < 8; ++v) {
            int tl = v + 8 * hl;
            fxs[wave][(nl)      * 17 + tl] = d0[v];
            fxs[wave][(16 + nl) * 17 + tl] = d1[v];
        }

        // ---- sequential recurrence: lane = row, 16 steps ----
        #pragma unroll
        for (int tt = 0; tt < 16; ++tt) {
            float fx = fxs[wave][lane * 17 + tt] * amp;
            float dz = (AZ_F * (bz_c * (goal - y) - z) + fx) * (1.0f / TAU_F);
            y += z * dy_c;        // uses old z
            z += dz * DT_F;
            yst[wave][lane * 16 + tt] = y;
        }

        // ---- coalesced store: 2 rows x 16 contiguous t per instruction ----
        #pragma unroll
        for (int pr = 0; pr < 16; ++pr) {
            int r  = pr * 2 + hl;
            int t  = base + 1 + nl;
            if (t < TSTEPS) {
                int rb = batch0 + (r >> 1);
                out[(size_t)rb * (2 * TSTEPS) + (r & 1) * TSTEPS + t] =
                    yst[wave][r * 16 + nl];
            }
        }
    }
}

// ---------------------------------------------------------------
extern "C" void kernel_launch(void* const* d_in, const int* in_sizes, int n_in,
                              void* d_out, int out_size, void* d_ws, size_t ws_size,
                              hipStream_t stream) {
    const float* x    = (const float*)d_in[0];   // (65536, 54)
    const float* W    = (const float*)d_in[1];   // (54, 54)
    const float* bias = (const float*)d_in[2];   // (54,)
    float* out = (float*)d_out;                  // (65536, 2, 301)
    float* Q   = (float*)d_ws;                   // QROWS * QSTRIDE floats

    dmp_q_kernel<<<1, QROWS, 0, stream>>>(Q);
    dmp_main_kernel<<<BATCH / 64, 128, 0, stream>>>(x, W, bias, Q, out);
}